// QMLP_14654428414264
// MI455X (gfx1250) — compile-verified
//
#include <hip/hip_runtime.h>
#include <math.h>

#define NQ    9
#define NST   512          // 2^9 amplitudes
#define NL    10
#define NE    5
#define NB    4096
#define NCIRC (NB*NE)      // 20480 independent circuits
#define NCLS  10
#define MT    16           // WMMA M tile (circuits per workgroup)
#define KCH   256          // K chunk resident in LDS
#define ASTR  (KCH+4)      // padded LDS stride (260: lanes land on distinct banks, 8B aligned)

typedef float v2f __attribute__((ext_vector_type(2)));
typedef float v8f __attribute__((ext_vector_type(8)));

// ---------------------------------------------------------------------------
// Kernel 1: build the 512x512 layer unitary, column by column.
// Block c propagates basis state |c> through 10 layers of (9 Rot + 9 CRX).
// Qubit i <-> bit (8-i) of the flat index (row-major (2,)*9 reshape).
// ---------------------------------------------------------------------------
__global__ void __launch_bounds__(256)
build_unitary(const float* __restrict__ rot_w, const float* __restrict__ crx_w,
              float* __restrict__ Ure, float* __restrict__ Uim) {
  __shared__ float pr[NST], pi_[NST];
  __shared__ float rotm[NL*NQ][8];   // u00r,u00i,u01r,u01i,u10r,u10i,u11r,u11i
  __shared__ float crxm[NL*NQ][2];   // cos(t/2), sin(t/2)
  const int t   = threadIdx.x;
  const int col = blockIdx.x;

  // Precompute all gate matrices once per block.
  if (t < NL*NQ) {
    float phi = rot_w[t*3+0], th = rot_w[t*3+1], om = rot_w[t*3+2];
    float sth, cth; sincosf(0.5f*th, &sth, &cth);
    float sa, ca;   sincosf(0.5f*(phi+om), &sa, &ca);
    float sb, cb;   sincosf(0.5f*(phi-om), &sb, &cb);
    // Rot = RZ(om) RY(th) RZ(phi)
    rotm[t][0] =  cth*ca; rotm[t][1] = -cth*sa;   // u00 = cth e^{-i(phi+om)/2}
    rotm[t][2] = -sth*cb; rotm[t][3] = -sth*sb;   // u01 = -sth e^{+i(phi-om)/2}
    rotm[t][4] =  sth*cb; rotm[t][5] = -sth*sb;   // u10 =  sth e^{-i(phi-om)/2}
    rotm[t][6] =  cth*ca; rotm[t][7] =  cth*sa;   // u11 =  cth e^{+i(phi+om)/2}
  } else if (t < 2*NL*NQ) {
    int g = t - NL*NQ;
    float s, c; sincosf(0.5f*crx_w[g], &s, &c);
    crxm[g][0] = c; crxm[g][1] = s;
  }
  for (int idx = t; idx < NST; idx += 256) { pr[idx] = (idx == col) ? 1.f : 0.f; pi_[idx] = 0.f; }
  __syncthreads();

  for (int n = 0; n < NL; ++n) {
    // 9 general single-qubit Rot gates; 256 disjoint amplitude pairs, 1/thread.
    for (int i = 0; i < NQ; ++i) {
      const float* u = rotm[n*NQ + i];
      const int b = 8 - i, mask = 1 << b;
      const int n0 = ((t >> b) << (b+1)) | (t & (mask-1));
      const int n1 = n0 | mask;
      float a0r = pr[n0], a0i = pi_[n0], a1r = pr[n1], a1i = pi_[n1];
      pr[n0]  = u[0]*a0r - u[1]*a0i + u[2]*a1r - u[3]*a1i;
      pi_[n0] = u[0]*a0i + u[1]*a0r + u[2]*a1i + u[3]*a1r;
      pr[n1]  = u[4]*a0r - u[5]*a0i + u[6]*a1r - u[7]*a1i;
      pi_[n1] = u[4]*a0i + u[5]*a0r + u[6]*a1i + u[7]*a1r;
      __syncthreads();
    }
    // 9 CRX gates on the ring: active only on control=1 subspace (128 pairs).
    for (int i = 0; i < NQ; ++i) {
      const float c = crxm[n*NQ + i][0], s = crxm[n*NQ + i][1];
      const int bc = 8 - i, bt = 8 - ((i + 1) % NQ);
      if (t < 128) {
        const int blo = min(bc, bt), bhi = max(bc, bt);
        int xx = t;
        xx = ((xx >> blo) << (blo+1)) | (xx & ((1 << blo) - 1));
        xx = ((xx >> bhi) << (bhi+1)) | (xx & ((1 << bhi) - 1));
        const int n0 = xx | (1 << bc);
        const int n1 = n0 | (1 << bt);
        float a0r = pr[n0], a0i = pi_[n0], a1r = pr[n1], a1i = pi_[n1];
        // RX on target: new0 = c*a0 - i s*a1 ; new1 = -i s*a0 + c*a1
        pr[n0]  =  c*a0r + s*a1i;  pi_[n0] =  c*a0i - s*a1r;
        pr[n1]  =  s*a0i + c*a1r;  pi_[n1] = -s*a0r + c*a1i;
      }
      __syncthreads();
    }
  }
  for (int idx = t; idx < NST; idx += 256) {
    Ure[idx*NST + col] = pr[idx];
    Uim[idx*NST + col] = pi_[idx];
  }
}

// ---------------------------------------------------------------------------
// Kernel 2: fused encode + complex GEMM (WMMA f32 16x16x4) + <Z> reduction.
// psi_out[m, n] = sum_k U[n, k] * psi_enc[m, k]; never materialized: reduced
// to z[m][qubit] = sum_n sign * |psi_out|^2 in registers/LDS.
// Fragment layout (ISA 7.12.2): lane&15 = M (A) / N (B) index, lane>>4 picks
// the K-half, VGPR pair = consecutive K.  C/D: VGPR j -> M = j + 8*(lane>>4).
// ---------------------------------------------------------------------------
__global__ void __launch_bounds__(256)
qgemm(const float* __restrict__ x, const float* __restrict__ rw,
      const float* __restrict__ Ure, const float* __restrict__ Uim,
      float* __restrict__ zbuf) {
  __shared__ float aRe[MT*ASTR], aIm[MT*ASTR];
  __shared__ float venc[MT][NQ][4];   // {re0,im0,re1,im1} per (circuit,qubit)
  __shared__ float zacc[MT][NQ];
  const int t     = threadIdx.x;
  const int lane  = t & 31;
  const int wave  = t >> 5;
  const int mtile = blockIdx.x;

  // Encoded per-qubit 2-vectors: v = RZ(t2) RY(t1) RX(t0) |0>
  if (t < MT*NQ) {
    const int mi = t / NQ, qi = t % NQ;
    const int m = mtile*MT + mi;
    const int b = m / NE, e = m % NE;
    const float xv = x[b*NQ + qi];
    const float* w = &rw[((b*NE + e)*NQ + qi)*3];
    float s0, c0, s1, c1, s2, c2;
    sincosf(0.5f*w[0]*xv, &s0, &c0);
    sincosf(0.5f*w[1]*xv, &s1, &c1);
    sincosf(0.5f*w[2]*xv, &s2, &c2);
    const float b0r = c1*c0, b0i = s1*s0, b1r = s1*c0, b1i = -c1*s0;
    venc[mi][qi][0] = c2*b0r + s2*b0i;   // v0 = e^{-i t2/2} * b0
    venc[mi][qi][1] = c2*b0i - s2*b0r;
    venc[mi][qi][2] = c2*b1r - s2*b1i;   // v1 = e^{+i t2/2} * b1
    venc[mi][qi][3] = c2*b1i + s2*b1r;
    zacc[mi][qi] = 0.f;
  }

  v8f cre[4], cim[4];
  for (int tt = 0; tt < 4; ++tt)
    for (int q = 0; q < 8; ++q) { cre[tt][q] = 0.f; cim[tt][q] = 0.f; }

  const int mi_l = lane & 15;
  const int koff = (lane >> 4) * 2;

  for (int kc = 0; kc < 2; ++kc) {
    const int kbase = kc * KCH;
    __syncthreads();                       // venc ready / previous chunk consumed
    // Build psi_enc K-chunk in LDS: Kronecker product of 9 2-vectors.
    {
      const int mi  = t & 15;
      const int nb0 = (t >> 4) * 16;       // 16 amplitudes per thread
      for (int j = 0; j < 16; ++j) {
        const int nl = nb0 + j;
        const int n  = kbase + nl;
        int bb = (n >> 8) & 1;
        float prr = venc[mi][0][bb*2+0], pri = venc[mi][0][bb*2+1];
        #pragma unroll
        for (int qi = 1; qi < NQ; ++qi) {
          bb = (n >> (8 - qi)) & 1;
          const float vr = venc[mi][qi][bb*2+0], vi = venc[mi][qi][bb*2+1];
          const float nr = prr*vr - pri*vi;
          pri = prr*vi + pri*vr; prr = nr;
        }
        aRe[mi*ASTR + nl] = prr; aIm[mi*ASTR + nl] = pri;
      }
    }
    __syncthreads();
    // Complex GEMM over this K chunk: 4 f32 WMMAs per K-step.
    for (int tt = 0; tt < 4; ++tt) {
      const int ntile = wave + tt*8;
      const int nrow  = ntile*16 + mi_l;
      const float* bre = &Ure[nrow*NST + kbase + koff];
      const float* bim = &Uim[nrow*NST + kbase + koff];
      #pragma unroll 4
      for (int kk = 0; kk < KCH; kk += 4) {
        v2f a_re = *(const v2f*)&aRe[mi_l*ASTR + kk + koff];
        v2f a_im = *(const v2f*)&aIm[mi_l*ASTR + kk + koff];
        v2f b_re = *(const v2f*)&bre[kk];
        v2f b_im = *(const v2f*)&bim[kk];
        v2f a_ni = -a_im;                  // f32 WMMA has no A/B negate modifier
        cre[tt] = __builtin_amdgcn_wmma_f32_16x16x4_f32(false, a_re, false, b_re, (short)0, cre[tt], false, false);
        cre[tt] = __builtin_amdgcn_wmma_f32_16x16x4_f32(false, a_ni, false, b_im, (short)0, cre[tt], false, false);
        cim[tt] = __builtin_amdgcn_wmma_f32_16x16x4_f32(false, a_re, false, b_im, (short)0, cim[tt], false, false);
        cim[tt] = __builtin_amdgcn_wmma_f32_16x16x4_f32(false, a_im, false, b_re, (short)0, cim[tt], false, false);
      }
    }
  }
  __syncthreads();

  // <Z_i> reduction. n = ntile*16 + nl: qubits 5..8 <-> bits 3..0 (lane-local),
  // qubits 0..4 <-> bits 4..0 of ntile (tile-constant signs).
  const int nl   = lane & 15;
  const int half = lane >> 4;
  for (int tt = 0; tt < 4; ++tt) {
    const int ntile = wave + tt*8;
    for (int j = 0; j < 8; ++j) {
      const float p = cre[tt][j]*cre[tt][j] + cim[tt][j]*cim[tt][j];
      float s0 = p;
      float s5 = ((nl >> 3) & 1) ? -p : p;
      float s6 = ((nl >> 2) & 1) ? -p : p;
      float s7 = ((nl >> 1) & 1) ? -p : p;
      float s8 = ( nl       & 1) ? -p : p;
      #pragma unroll
      for (int msk = 1; msk < 16; msk <<= 1) {
        s0 += __shfl_xor(s0, msk, 32);
        s5 += __shfl_xor(s5, msk, 32);
        s6 += __shfl_xor(s6, msk, 32);
        s7 += __shfl_xor(s7, msk, 32);
        s8 += __shfl_xor(s8, msk, 32);
      }
      if (nl == 0) {
        const int m = j + half*8;
        for (int qi = 0; qi < 5; ++qi) {
          const float sg = ((ntile >> (4 - qi)) & 1) ? -s0 : s0;
          atomicAdd(&zacc[m][qi], sg);
        }
        atomicAdd(&zacc[m][5], s5);
        atomicAdd(&zacc[m][6], s6);
        atomicAdd(&zacc[m][7], s7);
        atomicAdd(&zacc[m][8], s8);
      }
    }
  }
  __syncthreads();
  if (t < MT*NQ) {
    const int mi = t / NQ, qi = t % NQ;
    zbuf[(mtile*MT + mi)*NQ + qi] = zacc[mi][qi];
  }
}

// ---------------------------------------------------------------------------
// Kernel 3: mean over encodings, linear head, log_softmax.
// ---------------------------------------------------------------------------
__global__ void __launch_bounds__(256)
head_kernel(const float* __restrict__ zbuf, const float* __restrict__ fcw,
            const float* __restrict__ fcb, float* __restrict__ out) {
  const int b = blockIdx.x*blockDim.x + threadIdx.x;
  if (b >= NB) return;
  float q[NQ];
  #pragma unroll
  for (int i = 0; i < NQ; ++i) {
    float s = 0.f;
    for (int e = 0; e < NE; ++e) s += zbuf[(b*NE + e)*NQ + i];
    q[i] = s * (1.0f / NE);
  }
  float lg[NCLS], mx = -1e30f;
  #pragma unroll
  for (int j = 0; j < NCLS; ++j) {
    float s = fcb[j];
    for (int i = 0; i < NQ; ++i) s += fcw[j*NQ + i] * q[i];
    lg[j] = s; mx = fmaxf(mx, s);
  }
  float se = 0.f;
  for (int j = 0; j < NCLS; ++j) se += expf(lg[j] - mx);
  const float lse = logf(se);
  for (int j = 0; j < NCLS; ++j) out[b*NCLS + j] = lg[j] - mx - lse;
}

extern "C" void kernel_launch(void* const* d_in, const int* in_sizes, int n_in,
                              void* d_out, int out_size, void* d_ws, size_t ws_size,
                              hipStream_t stream) {
  (void)in_sizes; (void)n_in; (void)out_size; (void)ws_size;
  const float* x   = (const float*)d_in[0];
  const float* rw  = (const float*)d_in[1];
  const float* rot = (const float*)d_in[2];
  const float* crx = (const float*)d_in[3];
  const float* fcw = (const float*)d_in[4];
  const float* fcb = (const float*)d_in[5];
  float* Ure  = (float*)d_ws;                 // 512*512 f32
  float* Uim  = Ure + NST*NST;                // 512*512 f32
  float* zbuf = Uim + NST*NST;                // 20480*9 f32
  build_unitary<<<NST, 256, 0, stream>>>(rot, crx, Ure, Uim);
  qgemm<<<NCIRC/MT, 256, 0, stream>>>(x, rw, Ure, Uim, zbuf);
  head_kernel<<<(NB + 255)/256, 256, 0, stream>>>(zbuf, fcw, fcb, (float*)d_out);
}